// PCEN_1322849927828
// MI455X (gfx1250) — compile-verified
//
#include <hip/hip_runtime.h>
#include <hip/hip_bf16.h>

typedef __attribute__((ext_vector_type(2))) float v2f;
typedef __attribute__((ext_vector_type(8))) float v8f;

#define T_DIM 4096
#define F_DIM 128
// s = (sqrt(1+4*t^2)-1)/(2*t^2), t = 256
#define S_COEF 0.0038986280560f
#define EPS_C  1e-5f

__global__ __launch_bounds__(128) void pcen_wmma_kernel(
    const float* __restrict__ x,
    const float* __restrict__ alpha,
    const float* __restrict__ delta,
    const float* __restrict__ rpar,
    float* __restrict__ out)
{
  const int lane = threadIdx.x & 31;
  const int wave = blockIdx.x * (blockDim.x >> 5) + (threadIdx.x >> 5);
  const int n    = lane & 15;      // sequence slot (N dim) within the wave
  const int hi   = lane >> 4;      // 0: lanes 0-15, 1: lanes 16-31
  const int row  = wave * 16 + n;  // global row in [0, 4096) = b*128 + f
  const int f    = row & (F_DIM - 1);
  const int bidx = row >> 7;

  const float one_m_s = 1.0f - S_COEF;

  // per-row PCEN constants
  const float a_c  = __expf(alpha[f]);
  const float d_c  = __expf(delta[f]);
  const float rr_c = __expf(rpar[f]);
  const float drr  = __expf(rr_c * __logf(d_c));   // d^r

  // A operand: blocked-scan triangular matrix, A[m][j] = s*(1-s)^(m-j) for j<=m.
  // 16x16x4 f32 A layout: lanes 0-15 & 16-31 both carry M=0..15;
  // elem0 = K = 4s+2*hi, elem1 = K = 4s+2*hi+1.
  v2f A0, A1, A2, A3;
  {
    const int m = n;
#define COEF(J) (((J) <= m) ? (S_COEF * __powf(one_m_s, (float)(m - (J)))) : 0.0f)
    A0.x = COEF(0  + 2 * hi);  A0.y = COEF(1  + 2 * hi);
    A1.x = COEF(4  + 2 * hi);  A1.y = COEF(5  + 2 * hi);
    A2.x = COEF(8  + 2 * hi);  A2.y = COEF(9  + 2 * hi);
    A3.x = COEF(12 + 2 * hi);  A3.y = COEF(13 + 2 * hi);
#undef COEF
  }

  // carry decay (1-s)^(t_local+1) for this lane's 8 time slots (t_local = v + 8*hi)
  float cvec[8];
  {
    float cv = __powf(one_m_s, (float)(8 * hi + 1));
#pragma unroll
    for (int v = 0; v < 8; ++v) { cvec[v] = cv; cv *= one_m_s; }
  }

  const float* xrow = x + (size_t)row * T_DIM;
  float carry = xrow[0];  // virtual m_{-1} = x_0 so that m_0 = x_0 exactly

  for (int tb = 0; tb < T_DIM / 16; ++tb) {
    const int t0 = tb * 16;
    // full 16-time tile of this lane's row, register resident
    const float4 r0 = *reinterpret_cast<const float4*>(xrow + t0 + 0);
    const float4 r1 = *reinterpret_cast<const float4*>(xrow + t0 + 4);
    const float4 r2 = *reinterpret_cast<const float4*>(xrow + t0 + 8);
    const float4 r3 = *reinterpret_cast<const float4*>(xrow + t0 + 12);

    if (t0 + 16 < T_DIM) __builtin_prefetch(xrow + t0 + 16, 0, 0);

    // B operand (4x16): B[k][n] = x[row n][t0 + 4s + 2*hi + {0,1}]
    v2f B0, B1, B2, B3;
    B0.x = hi ? r0.z : r0.x;  B0.y = hi ? r0.w : r0.y;
    B1.x = hi ? r1.z : r1.x;  B1.y = hi ? r1.w : r1.y;
    B2.x = hi ? r2.z : r2.x;  B2.y = hi ? r2.w : r2.y;
    B3.x = hi ? r3.z : r3.x;  B3.y = hi ? r3.w : r3.y;

    // D[m][n] = sum_j A[m][j] * x[t0+j][n]  via 4 chained K=4 fp32 WMMAs
    v8f acc = {};
    acc = __builtin_amdgcn_wmma_f32_16x16x4_f32(false, A0, false, B0, (short)0, acc, false, false);
    acc = __builtin_amdgcn_wmma_f32_16x16x4_f32(false, A1, false, B1, (short)0, acc, false, false);
    acc = __builtin_amdgcn_wmma_f32_16x16x4_f32(false, A2, false, B2, (short)0, acc, false, false);
    acc = __builtin_amdgcn_wmma_f32_16x16x4_f32(false, A3, false, B3, (short)0, acc, false, false);

    // add decayed carry -> true smoother values m[t0 + v + 8*hi]
    float mv[8];
#pragma unroll
    for (int v = 0; v < 8; ++v) mv[v] = acc[v] + cvec[v] * carry;

    // next-block carry = m[t0+15], held in elem 7 of the high-half lane n+16
    carry = __int_as_float(
        __builtin_amdgcn_ds_bpermute((n + 16) << 2, __float_as_int(mv[7])));

    // pointwise PCEN + transposed store out[b][t][f]
    float xl[8] = {r0.x, r0.y, r0.z, r0.w, r1.x, r1.y, r1.z, r1.w};
    float xh[8] = {r2.x, r2.y, r2.z, r2.w, r3.x, r3.y, r3.z, r3.w};
#pragma unroll
    for (int v = 0; v < 8; ++v) {
      const float xv     = hi ? xh[v] : xl[v];
      // smooth = (eps + m)^(-alpha) == exp(-a * log(eps + m))
      const float smooth = __expf(-a_c * __logf(EPS_C + mv[v]));
      const float pcen   = __expf(rr_c * __logf(xv * smooth + d_c)) - drr;
      const int   t      = t0 + v + 8 * hi;
      out[((size_t)bidx * T_DIM + t) * F_DIM + f] = pcen;
    }
  }
}

extern "C" void kernel_launch(void* const* d_in, const int* in_sizes, int n_in,
                              void* d_out, int out_size, void* d_ws, size_t ws_size,
                              hipStream_t stream) {
  const float* x     = (const float*)d_in[0];
  const float* alpha = (const float*)d_in[1];
  const float* delta = (const float*)d_in[2];
  const float* r     = (const float*)d_in[3];
  float* out = (float*)d_out;

  // 4096 rows / 16 rows-per-wave = 256 waves = 64 blocks x 4 waves
  dim3 grid(64), block(128);
  pcen_wmma_kernel<<<grid, block, 0, stream>>>(x, alpha, delta, r, out);
}